// MultiGAT_57621281243371
// MI455X (gfx1250) — compile-verified
//
#include <hip/hip_runtime.h>
#include <hip/hip_bf16.h>

#define N_IN_DIM 128
#define N_HID    128
#define N_HEADS  8
#define N_OUT    64
#define LRELU    0.2f

typedef __attribute__((ext_vector_type(2))) float v2f;
typedef __attribute__((ext_vector_type(4))) float v4f;
typedef __attribute__((ext_vector_type(8))) float v8f;

// ---------------------------------------------------------------------------
// fp32 WMMA GEMM:  C[M x NCOLS] = A[M x 128] * B[128 x NCOLS]   (NCOLS const)
// One block: 16 rows, NCOLS/16 waves, each wave one 16x16 tile, K=128.
// A tile staged in LDS (float4 staging); fragment layout per CDNA5 16x16x4 f32:
//   A (16x4): lane L -> row m=L%16, VGPR0=K(2*(L/16)), VGPR1=K(2*(L/16)+1)
//   B (4x16): lane L -> col n=L%16, VGPR0=K(2*(L/16)), VGPR1=K(2*(L/16)+1)
//   C (16x16): VGPR r -> row r + 8*(L/16), col L%16
// With NCOLS constexpr, all B/C addressing folds into 24-bit imm offsets.
// ---------------------------------------------------------------------------
template <int NCOLS>
__global__ void gemm_wmma_f32(const float* __restrict__ A,
                              const float* __restrict__ B,
                              float* __restrict__ C) {
  __shared__ float As[16 * 128];
  const int tid = threadIdx.x;
  const long long m0 = (long long)blockIdx.x * 16;

  // stage A tile (16x128 f32 = 512 float4) into LDS
  {
    const v4f* Ag = (const v4f*)(A + m0 * 128);
    v4f* Al = (v4f*)As;
    for (int idx = tid; idx < 16 * 32; idx += blockDim.x) Al[idx] = Ag[idx];
  }
  __syncthreads();

  const int lane = tid & 31;
  const int nloc = lane & 15;
  const int half = lane >> 4;                 // 0 or 1
  const int kb   = half * 2;
  const int n    = (tid >> 5) * 16 + nloc;    // global output column

  const float* Brow = B + (long long)kb * NCOLS + n;   // base; k0 via imm offset
  const float* Arow = As + nloc * 128 + kb;

  v8f acc = {0.f, 0.f, 0.f, 0.f, 0.f, 0.f, 0.f, 0.f};
#pragma unroll
  for (int k0 = 0; k0 < 128; k0 += 4) {
    v2f a, b;
    a.x = Arow[k0];
    a.y = Arow[k0 + 1];
    b.x = Brow[(long long)k0 * NCOLS];
    b.y = Brow[(long long)(k0 + 1) * NCOLS];
    acc = __builtin_amdgcn_wmma_f32_16x16x4_f32(false, a, false, b,
                                                (short)0, acc, false, false);
  }
  float* Crow = C + (m0 + half * 8) * NCOLS + n;
#pragma unroll
  for (int r = 0; r < 8; ++r) Crow[(long long)r * NCOLS] = acc[r];
}

// ---------------------------------------------------------------------------
// helpers
// ---------------------------------------------------------------------------
__device__ __forceinline__ void edge_sd(const int* __restrict__ ei, long long e,
                                        int Eorig, int& s, int& d) {
  if (e < Eorig) { s = ei[e]; d = ei[(long long)Eorig + e]; }
  else           { s = d = (int)(e - Eorig); }              // self loop
}

__device__ __forceinline__ void atomicMaxF(float* addr, float v) {
  if (v >= 0.f) atomicMax((int*)addr, __float_as_int(v));
  else          atomicMin((unsigned int*)addr, __float_as_uint(v));
}

__device__ __forceinline__ float lrelu(float v) { return v > 0.f ? v : LRELU * v; }

__global__ void fill_kernel(float* p, long long n, float v) {
  long long t = (long long)blockIdx.x * blockDim.x + threadIdx.x;
  if (t < n) p[t] = v;
}

// ---------------------------------------------------------------------------
// attention coefficients
// ---------------------------------------------------------------------------
__global__ void alpha_mh_kernel(const float* __restrict__ h,
                                const float* __restrict__ a_src,
                                const float* __restrict__ a_dst,
                                float* __restrict__ asv, float* __restrict__ adv,
                                int n) {  // per (node, head), C/H = 16
  long long t = (long long)blockIdx.x * blockDim.x + threadIdx.x;
  if (t >= (long long)n * N_HEADS) return;
  int i = (int)(t >> 3), hd = (int)(t & 7);
  const float* hp = h + (long long)i * N_HID + hd * 16;
  const float* a1 = a_src + hd * 16;
  const float* a2 = a_dst + hd * 16;
  float s1 = 0.f, s2 = 0.f;
#pragma unroll
  for (int c = 0; c < 16; ++c) { float v = hp[c]; s1 += v * a1[c]; s2 += v * a2[c]; }
  asv[t] = s1; adv[t] = s2;
}

__global__ void alpha_1h_kernel(const float* __restrict__ h,
                                const float* __restrict__ a_src,
                                const float* __restrict__ a_dst,
                                float* __restrict__ asv, float* __restrict__ adv,
                                int n) {  // per node, 64 channels, one head
  int i = blockIdx.x * blockDim.x + threadIdx.x;
  if (i >= n) return;
  const float* hp = h + (long long)i * N_OUT;
  float s1 = 0.f, s2 = 0.f;
#pragma unroll
  for (int c = 0; c < N_OUT; ++c) { float v = hp[c]; s1 += v * a_src[c]; s2 += v * a_dst[c]; }
  asv[i] = s1; adv[i] = s2;
}

// ---------------------------------------------------------------------------
// edge softmax passes (H, C compile-time -> index math is shifts/masks)
// ---------------------------------------------------------------------------
template <int H>
__global__ void edge_max_kernel(const int* __restrict__ ei, int Eorig, int Etot,
                                const float* __restrict__ asv,
                                const float* __restrict__ adv,
                                float* __restrict__ m) {
  long long t = (long long)blockIdx.x * blockDim.x + threadIdx.x;
  if (t >= (long long)Etot * H) return;
  long long e = t / H; int hd = (int)(t % H);
  int s, d; edge_sd(ei, e, Eorig, s, d);
  float v = lrelu(asv[(long long)s * H + hd] + adv[(long long)d * H + hd]);
  atomicMaxF(&m[(long long)d * H + hd], v);
}

template <int H>
__global__ void edge_sum_kernel(const int* __restrict__ ei, int Eorig, int Etot,
                                const float* __restrict__ asv,
                                const float* __restrict__ adv,
                                const float* __restrict__ m,
                                float* __restrict__ dn) {
  long long t = (long long)blockIdx.x * blockDim.x + threadIdx.x;
  if (t >= (long long)Etot * H) return;
  long long e = t / H; int hd = (int)(t % H);
  int s, d; edge_sd(ei, e, Eorig, s, d);
  long long di = (long long)d * H + hd;
  float v = lrelu(asv[(long long)s * H + hd] + adv[di]);
  atomicAdd(&dn[di], expf(v - m[di]));
}

template <int H, int C>
__global__ void edge_agg_kernel(const int* __restrict__ ei, int Eorig, int Etot,
                                const float* __restrict__ asv,
                                const float* __restrict__ adv,
                                const float* __restrict__ m,
                                const float* __restrict__ dn,
                                const float* __restrict__ hsrc,
                                float* __restrict__ out) {
  long long t = (long long)blockIdx.x * blockDim.x + threadIdx.x;
  if (t >= (long long)Etot * C) return;
  long long e = t / C; int c = (int)(t % C);
  int hd = c / (C / H);
  int s, d; edge_sd(ei, e, Eorig, s, d);
  long long di = (long long)d * H + hd;
  float v = lrelu(asv[(long long)s * H + hd] + adv[di]);
  float alpha = expf(v - m[di]) / (dn[di] + 1e-16f);
  atomicAdd(&out[(long long)d * C + c], hsrc[(long long)s * C + c] * alpha);
}

// ---------------------------------------------------------------------------
// pointwise: bias + ELU (layer 1 output)
// ---------------------------------------------------------------------------
__global__ void elu_bias_kernel(float* __restrict__ x, const float* __restrict__ b,
                                long long n) {
  long long t = (long long)blockIdx.x * blockDim.x + threadIdx.x;
  if (t >= n) return;
  float v = x[t] + b[(int)(t & (N_HID - 1))];
  x[t] = v > 0.f ? v : expm1f(v);
}

// ---------------------------------------------------------------------------
// final: out = log_softmax(agg2 + b2) over 64 channels
// ---------------------------------------------------------------------------
__global__ void logsoftmax_kernel(const float* __restrict__ agg,
                                  const float* __restrict__ b,
                                  float* __restrict__ out, int n) {
  int i = blockIdx.x * blockDim.x + threadIdx.x;
  if (i >= n) return;
  const float* p = agg + (long long)i * N_OUT;
  float mx = -__builtin_inff();
#pragma unroll
  for (int c = 0; c < N_OUT; ++c) mx = fmaxf(mx, p[c] + b[c]);
  float s = 0.f;
#pragma unroll
  for (int c = 0; c < N_OUT; ++c) s += expf(p[c] + b[c] - mx);
  float ls = logf(s) + mx;
  float* o = out + (long long)i * N_OUT;
#pragma unroll
  for (int c = 0; c < N_OUT; ++c) o[c] = p[c] + b[c] - mx - ls;
}

// ---------------------------------------------------------------------------
// host launcher
// ---------------------------------------------------------------------------
static inline unsigned nblk(long long n) { return (unsigned)((n + 255) / 256); }

extern "C" void kernel_launch(void* const* d_in, const int* in_sizes, int n_in,
                              void* d_out, int out_size, void* d_ws, size_t ws_size,
                              hipStream_t stream) {
  const float* x   = (const float*)d_in[0];
  const int*   ei  = (const int*)  d_in[1];
  const float* W1  = (const float*)d_in[2];
  const float* as1 = (const float*)d_in[3];
  const float* ad1 = (const float*)d_in[4];
  const float* b1  = (const float*)d_in[5];
  const float* W2  = (const float*)d_in[6];
  const float* as2 = (const float*)d_in[7];
  const float* ad2 = (const float*)d_in[8];
  const float* b2  = (const float*)d_in[9];
  float* out = (float*)d_out;

  const int N = in_sizes[0] / N_IN_DIM;   // 50000 (multiple of 16)
  const int E = in_sizes[1] / 2;          // 800000
  const int Etot = E + N;                 // self loops appended

  // ---- workspace layout (floats) ----
  float* p = (float*)d_ws;
  float* h_lin1 = p;                 p += (long long)N * N_HID;   // x @ W1
  float* as1v   = p;                 p += (long long)N * N_HEADS;
  float* ad1v   = p;                 p += (long long)N * N_HEADS;
  float* m1     = p;                 p += (long long)N * N_HEADS; // -inf region start
  float* m2     = p;                 p += (long long)N;           // -inf region end
  float* d1     = p;                 p += (long long)N * N_HEADS; // zero region start
  float* agg1   = p;                 p += (long long)N * N_HID;   //   (becomes h after ELU)
  float* d2     = p;                 p += (long long)N;
  float* agg2   = p;                 p += (long long)N * N_OUT;   // zero region end
  float* h_lin2 = p;                 p += (long long)N * N_OUT;   // h @ W2
  float* as2v   = p;                 p += (long long)N;
  float* ad2v   = p;                 /* end */

  const long long neginf_n = (long long)N * (N_HEADS + 1);                 // m1+m2
  const long long zero_n   = (long long)N * (N_HEADS + N_HID + 1 + N_OUT); // d1+agg1+d2+agg2

  // re-init every call (graph-replay safe)
  hipMemsetAsync(d1, 0, zero_n * sizeof(float), stream);
  fill_kernel<<<nblk(neginf_n), 256, 0, stream>>>(m1, neginf_n, -__builtin_inff());

  // ================= layer 1 (8 heads x 16 ch, concat) =================
  gemm_wmma_f32<N_HID><<<N / 16, (N_HID / 16) * 32, 0, stream>>>(x, W1, h_lin1);
  alpha_mh_kernel<<<nblk((long long)N * N_HEADS), 256, 0, stream>>>(
      h_lin1, as1, ad1, as1v, ad1v, N);
  edge_max_kernel<N_HEADS><<<nblk((long long)Etot * N_HEADS), 256, 0, stream>>>(
      ei, E, Etot, as1v, ad1v, m1);
  edge_sum_kernel<N_HEADS><<<nblk((long long)Etot * N_HEADS), 256, 0, stream>>>(
      ei, E, Etot, as1v, ad1v, m1, d1);
  edge_agg_kernel<N_HEADS, N_HID><<<nblk((long long)Etot * N_HID), 256, 0, stream>>>(
      ei, E, Etot, as1v, ad1v, m1, d1, h_lin1, agg1);
  elu_bias_kernel<<<nblk((long long)N * N_HID), 256, 0, stream>>>(
      agg1, b1, (long long)N * N_HID);

  // ================= layer 2 (1 head x 64 ch, mean==identity) =================
  gemm_wmma_f32<N_OUT><<<N / 16, (N_OUT / 16) * 32, 0, stream>>>(agg1, W2, h_lin2);
  alpha_1h_kernel<<<nblk(N), 256, 0, stream>>>(h_lin2, as2, ad2, as2v, ad2v, N);
  edge_max_kernel<1><<<nblk(Etot), 256, 0, stream>>>(ei, E, Etot, as2v, ad2v, m2);
  edge_sum_kernel<1><<<nblk(Etot), 256, 0, stream>>>(ei, E, Etot, as2v, ad2v, m2, d2);
  edge_agg_kernel<1, N_OUT><<<nblk((long long)Etot * N_OUT), 256, 0, stream>>>(
      ei, E, Etot, as2v, ad2v, m2, d2, h_lin2, agg2);

  logsoftmax_kernel<<<nblk(N), 256, 0, stream>>>(agg2, b2, out, N);
}